// LeNet_5_filocl_10161892622476
// MI455X (gfx1250) — compile-verified
//
#include <hip/hip_runtime.h>
#include <math.h>

// ---------------------------------------------------------------------------
// LeNet-5 forward on gfx1250 (MI455X), fp32 end-to-end.
// All conv/FC layers are implicit GEMMs on V_WMMA_F32_16X16X4_F32 (exact fp32
// on the matrix pipe; the net is memory-bound at ~21 GFLOP / ~400 MB so low
// precision WMMA buys nothing). One wave32 per 16x16 output tile.
// LDS operands are built zero-padded im2col style so the K-loop is just
// two aligned ds_load_b64 + v_wmma, with EXEC all-1s throughout.
// LDS staging uses gfx1250 async global->LDS loads when available.
// ---------------------------------------------------------------------------

typedef __attribute__((ext_vector_type(2))) float v2f;
typedef __attribute__((ext_vector_type(8))) float v8f;

#if defined(__has_builtin)
#if __has_builtin(__builtin_amdgcn_global_load_async_to_lds_b32) && \
    __has_builtin(__builtin_amdgcn_s_wait_asynccnt)
#define LENET_HAVE_ASYNC_LDS 1
#endif
#endif
#ifndef LENET_HAVE_ASYNC_LDS
#define LENET_HAVE_ASYNC_LDS 0
#endif

// Stage one dword global -> LDS (async on gfx1250, sync fallback otherwise).
static __device__ __forceinline__ void lds_fill(float* lptr, const float* gptr) {
#if LENET_HAVE_ASYNC_LDS
    __builtin_amdgcn_global_load_async_to_lds_b32(
        (__attribute__((address_space(1))) int*)gptr,
        (__attribute__((address_space(3))) int*)lptr, 0, 0);
#else
    *lptr = *gptr;
#endif
}
static __device__ __forceinline__ void lds_fence() {
#if LENET_HAVE_ASYNC_LDS
    __builtin_amdgcn_s_wait_asynccnt(0);
#endif
    __syncthreads();
}

// D = A(16x4) * B(4x16) + C, fp32.
// A frag: lane L holds row M=L&15; VGPR0/1 = K+{0,1} (lanes 0-15) or K+{2,3} (16-31).
// B frag: lane L holds col N=L&15; same K split.
// C/D:    lane L holds col N=L&15; VGPR r = row M = r + 8*(L>>4).
static __device__ __forceinline__ v8f wmma_f32(v2f a, v2f b, v8f c) {
    return __builtin_amdgcn_wmma_f32_16x16x4_f32(false, a, false, b, (short)0, c,
                                                 false, false);
}

// Branchless tanh: sign(x) * (1 - 2/(e^{2|x|}+1)); e->inf saturates to 1.
static __device__ __forceinline__ float fast_tanh(float x) {
    float ax = __builtin_fabsf(x);
    float e  = __expf(2.0f * ax);                   // v_exp_f32 path
    float r  = __builtin_amdgcn_rcpf(e + 1.0f);     // v_rcp_f32
    float t  = 1.0f - 2.0f * r;
    return __builtin_copysignf(t, x);
}

// ---------------------------------------------------------------------------
// conv0 (WMMA): x[B,1,32,32] -> a0[B,16,32,32], 5x5 SAME + tanh.
// GEMM: M=16 filters, K=25 (pad 28), N tile = 16 x-positions. Grid = B*32*2.
// ---------------------------------------------------------------------------
__global__ __launch_bounds__(32) void k_conv0_wmma(const float* __restrict__ x,
                                                   const float* __restrict__ filt,
                                                   float* __restrict__ a0) {
    __shared__ float sw[16 * 28]; // [m][K=28], K>=25 zero
    __shared__ float sp[16 * 28]; // [n][K=28] im2col, OOB/pad zero
    int t  = blockIdx.x;
    int xt = t & 1;  t >>= 1;
    int y  = t & 31;
    int b  = t >> 5;
    int x0 = xt * 16;
    int lane = threadIdx.x;

    for (int i = lane; i < 448; i += 32) {
        int m = i / 28, k = i % 28;
        if (k < 25) lds_fill(&sw[i], &filt[m * 25 + k]); else sw[i] = 0.f;
    }
    for (int i = lane; i < 448; i += 32) {
        int n = i / 28, k = i % 28;
        float zero_needed = 1.f;
        if (k < 25) {
            int dy = k / 5, dx = k % 5;
            int iy = y + dy - 2, ix = x0 + n + dx - 2;
            if (iy >= 0 && iy < 32 && ix >= 0 && ix < 32) {
                lds_fill(&sp[i], &x[(b * 32 + iy) * 32 + ix]);
                zero_needed = 0.f;
            }
        }
        if (zero_needed != 0.f) sp[i] = 0.f;
    }
    lds_fence();

    int lid = lane & 15, kof = (lane >> 4) * 2, half = lane >> 4;
    v8f acc = {};
#pragma unroll
    for (int kk = 0; kk < 28; kk += 4) {
        v2f av = *(const v2f*)&sw[lid * 28 + kk + kof];
        v2f bv = *(const v2f*)&sp[lid * 28 + kk + kof];
        acc = wmma_f32(av, bv, acc);
    }
#pragma unroll
    for (int r = 0; r < 8; ++r) {
        int m = r + 8 * half;
        a0[((b * 16 + m) * 32 + y) * 32 + x0 + lid] = fast_tanh(acc[r]);
    }
}

// ---------------------------------------------------------------------------
// conv1 (WMMA): a0[B,16,32,32] -> a1[B,6,28,28], 5x5 VALID + bias + tanh.
// GEMM: M=6 (pad 16), K=400, N tile = 16 x-positions. Grid = B*28*2.
// ---------------------------------------------------------------------------
__global__ __launch_bounds__(32) void k_conv1_wmma(const float* __restrict__ a0,
                                                   const float* __restrict__ w,
                                                   const float* __restrict__ bias,
                                                   float* __restrict__ a1) {
    __shared__ float sw[16 * 400]; // [m][400], rows 6..15 zero
    __shared__ float sp[16 * 400]; // [n][400] im2col (clamped; unused cols finite)
    int t  = blockIdx.x;
    int xt = t & 1;  t >>= 1;
    int y  = t % 28;
    int b  = t / 28;
    int x0 = xt * 16;
    int lane = threadIdx.x;

    for (int i = lane; i < 6400; i += 32) {
        if (i < 2400) lds_fill(&sw[i], &w[i]); else sw[i] = 0.f;
    }
    for (int i = lane; i < 6400; i += 32) {
        int n = i / 400, k = i % 400;
        int c = k / 25, r = k % 25;
        int dy = r / 5, dx = r % 5;
        int ix = x0 + n + dx;  ix = ix < 31 ? ix : 31; // clamp: cols >=28 unused
        lds_fill(&sp[i], &a0[((b * 16 + c) * 32 + (y + dy)) * 32 + ix]);
    }
    lds_fence();

    int lid = lane & 15, kof = (lane >> 4) * 2, half = lane >> 4;
    v8f acc = {};
#pragma unroll 2
    for (int kk = 0; kk < 400; kk += 4) {
        v2f av = *(const v2f*)&sw[lid * 400 + kk + kof];
        v2f bv = *(const v2f*)&sp[lid * 400 + kk + kof];
        acc = wmma_f32(av, bv, acc);
    }

    int xo = x0 + lid;
    if (xo < 28) {
#pragma unroll
        for (int r = 0; r < 8; ++r) {
            int m = r + 8 * half;
            if (m < 6)
                a1[((b * 6 + m) * 28 + y) * 28 + xo] = fast_tanh(acc[r] + bias[m]);
        }
    }
}

// ---------------------------------------------------------------------------
// 2x2 average pool, stride 2 (generic over C,Ho,Wo).
// ---------------------------------------------------------------------------
__global__ __launch_bounds__(256) void k_avgpool2(const float* __restrict__ in,
                                                  float* __restrict__ out,
                                                  int total, int C, int Ho, int Wo) {
    int i = blockIdx.x * blockDim.x + threadIdx.x;
    if (i >= total) return;
    int wo = i % Wo; int t = i / Wo;
    int ho = t % Ho; t /= Ho;
    int c  = t % C;  int b = t / C;
    int Wi = 2 * Wo;
    const float* p = in + ((size_t)(b * C + c) * (2 * Ho) + 2 * ho) * Wi + 2 * wo;
    out[i] = 0.25f * (p[0] + p[1] + p[Wi] + p[Wi + 1]);
}

// ---------------------------------------------------------------------------
// conv3 (WMMA): p1[B,6,14,14] -> a3[B,16,10,10], 5x5 VALID + bias + tanh.
// GEMM: M=16, K=150 (pad 152), N tile = one 10-wide output row. Grid = B*10.
// ---------------------------------------------------------------------------
__global__ __launch_bounds__(32) void k_conv3_wmma(const float* __restrict__ p1,
                                                   const float* __restrict__ w,
                                                   const float* __restrict__ bias,
                                                   float* __restrict__ a3) {
    __shared__ float sw[16 * 152]; // [m][152], k>=150 zero
    __shared__ float sp[16 * 152]; // [n][152] im2col, k>=150 zero, n>=10 clamped
    int t = blockIdx.x;
    int y = t % 10;
    int b = t / 10;
    int lane = threadIdx.x;

    for (int i = lane; i < 2432; i += 32) {
        int m = i / 152, k = i % 152;
        if (k < 150) lds_fill(&sw[i], &w[m * 150 + k]); else sw[i] = 0.f;
    }
    for (int i = lane; i < 2432; i += 32) {
        int n = i / 152, k = i % 152;
        if (k < 150) {
            int c = k / 25, r = k % 25;
            int dy = r / 5, dx = r % 5;
            int ix = n + dx;  ix = ix < 13 ? ix : 13; // clamp: cols >=10 unused
            lds_fill(&sp[i], &p1[((b * 6 + c) * 14 + (y + dy)) * 14 + ix]);
        } else {
            sp[i] = 0.f;
        }
    }
    lds_fence();

    int lid = lane & 15, kof = (lane >> 4) * 2, half = lane >> 4;
    v8f acc = {};
#pragma unroll 2
    for (int kk = 0; kk < 152; kk += 4) {
        v2f av = *(const v2f*)&sw[lid * 152 + kk + kof];
        v2f bv = *(const v2f*)&sp[lid * 152 + kk + kof];
        acc = wmma_f32(av, bv, acc);
    }

    if (lid < 10) {
#pragma unroll
        for (int r = 0; r < 8; ++r) {
            int m = r + 8 * half;
            a3[((b * 16 + m) * 10 + y) * 10 + lid] = fast_tanh(acc[r] + bias[m]);
        }
    }
}

// ---------------------------------------------------------------------------
// c5 (WMMA): p3[B,400] -> a5[B,120] = tanh(W[120,400].p3^T + b).
// Grid = (B/16) * 8 M-tiles; both operands staged in LDS.
// ---------------------------------------------------------------------------
__global__ __launch_bounds__(32) void k_c5_wmma(const float* __restrict__ p3,
                                                const float* __restrict__ w,
                                                const float* __restrict__ bias,
                                                float* __restrict__ a5) {
    __shared__ float sw[16 * 400]; // weight M-tile, rows >=120 zero
    __shared__ float sb[16 * 400]; // 16 images x 400
    int t  = blockIdx.x;
    int mt = t & 7;
    int b0 = (t >> 3) * 16;
    int m0 = mt * 16;
    int lane = threadIdx.x;

    for (int i = lane; i < 6400; i += 32) {
        int m = m0 + i / 400;
        if (m < 120) lds_fill(&sw[i], &w[m * 400 + i % 400]); else sw[i] = 0.f;
    }
    for (int i = lane; i < 6400; i += 32) lds_fill(&sb[i], &p3[b0 * 400 + i]);
    lds_fence();

    int lid = lane & 15, kof = (lane >> 4) * 2, half = lane >> 4;
    v8f acc = {};
#pragma unroll 2
    for (int kk = 0; kk < 400; kk += 4) {
        v2f av = *(const v2f*)&sw[lid * 400 + kk + kof];
        v2f bv = *(const v2f*)&sb[lid * 400 + kk + kof];
        acc = wmma_f32(av, bv, acc);
    }

    int n = b0 + lid;
#pragma unroll
    for (int r = 0; r < 8; ++r) {
        int m = m0 + r + 8 * half;
        if (m < 120) a5[n * 120 + m] = fast_tanh(acc[r] + bias[m]);
    }
}

// ---------------------------------------------------------------------------
// fc1 (WMMA): a5[B,120] -> a6[B,84] = tanh(a5.W^T + b), W=[84,120].
// Grid = (B/16) * 6 M-tiles. Weights in LDS, activations direct v2f global.
// ---------------------------------------------------------------------------
__global__ __launch_bounds__(32) void k_fc1_wmma(const float* __restrict__ a5,
                                                 const float* __restrict__ w,
                                                 const float* __restrict__ bias,
                                                 float* __restrict__ a6) {
    __shared__ float sw[16 * 120]; // rows >= 84 zero
    int t  = blockIdx.x;
    int mt = t % 6;
    int b0 = (t / 6) * 16;
    int m0 = mt * 16;
    int lane = threadIdx.x;

    for (int i = lane; i < 1920; i += 32) {
        int m = m0 + i / 120;
        if (m < 84) lds_fill(&sw[i], &w[m * 120 + i % 120]); else sw[i] = 0.f;
    }
    lds_fence();

    int lid = lane & 15, kof = (lane >> 4) * 2, half = lane >> 4;
    const float* brow = a5 + (b0 + lid) * 120;
    v8f acc = {};
#pragma unroll
    for (int kk = 0; kk < 120; kk += 4) {
        v2f av = *(const v2f*)&sw[lid * 120 + kk + kof];
        v2f bv = *(const v2f*)&brow[kk + kof];
        acc = wmma_f32(av, bv, acc);
    }

    int n = b0 + lid;
#pragma unroll
    for (int r = 0; r < 8; ++r) {
        int m = m0 + r + 8 * half;
        if (m < 84) a6[n * 84 + m] = fast_tanh(acc[r] + bias[m]);
    }
}

// ---------------------------------------------------------------------------
// fc2 (WMMA): a6[B,84] -> out[B,10] = a6.W^T + b, W=[10,84]. Grid = B/16.
// ---------------------------------------------------------------------------
__global__ __launch_bounds__(32) void k_fc2_wmma(const float* __restrict__ a6,
                                                 const float* __restrict__ w,
                                                 const float* __restrict__ bias,
                                                 float* __restrict__ out) {
    __shared__ float sw[16 * 84]; // rows >= 10 zero
    int b0 = blockIdx.x * 16;
    int lane = threadIdx.x;

    for (int i = lane; i < 1344; i += 32) {
        int m = i / 84;
        if (m < 10) lds_fill(&sw[i], &w[i]); else sw[i] = 0.f;
    }
    lds_fence();

    int lid = lane & 15, kof = (lane >> 4) * 2, half = lane >> 4;
    const float* brow = a6 + (b0 + lid) * 84;
    v8f acc = {};
#pragma unroll
    for (int kk = 0; kk < 84; kk += 4) {
        v2f av = *(const v2f*)&sw[lid * 84 + kk + kof];
        v2f bv = *(const v2f*)&brow[kk + kof];
        acc = wmma_f32(av, bv, acc);
    }

    int n = b0 + lid;
#pragma unroll
    for (int r = 0; r < 8; ++r) {
        int m = r + 8 * half;
        if (m < 10) out[n * 10 + m] = acc[r] + bias[m];
    }
}

// ---------------------------------------------------------------------------
// Host-side launcher.
// ---------------------------------------------------------------------------
extern "C" void kernel_launch(void* const* d_in, const int* in_sizes, int n_in,
                              void* d_out, int out_size, void* d_ws, size_t ws_size,
                              hipStream_t stream) {
    const float* x     = (const float*)d_in[0];
    const float* filt  = (const float*)d_in[1];
    const float* c1_w  = (const float*)d_in[2];
    const float* c1_b  = (const float*)d_in[3];
    const float* c3_w  = (const float*)d_in[4];
    const float* c3_b  = (const float*)d_in[5];
    const float* c5_w  = (const float*)d_in[6];
    const float* c5_b  = (const float*)d_in[7];
    const float* l1_w  = (const float*)d_in[8];
    const float* l1_b  = (const float*)d_in[9];
    const float* l2_w  = (const float*)d_in[10];
    const float* l2_b  = (const float*)d_in[11];
    float* out = (float*)d_out;

    const int B = in_sizes[0] / (32 * 32); // 4096

    // Workspace layout (floats)
    float* ws = (float*)d_ws;
    float* a0 = ws;                          // [B,16,32,32]
    float* a1 = a0 + (size_t)B * 16 * 1024;  // [B,6,28,28]
    float* p1 = a1 + (size_t)B * 6 * 784;    // [B,6,14,14]
    float* a3 = p1 + (size_t)B * 6 * 196;    // [B,16,10,10]
    float* p3 = a3 + (size_t)B * 16 * 100;   // [B,16,5,5] == [B,400]
    float* a5 = p3 + (size_t)B * 400;        // [B,120]
    float* a6 = a5 + (size_t)B * 120;        // [B,84]

    k_conv0_wmma<<<B * 32 * 2, 32, 0, stream>>>(x, filt, a0);
    k_conv1_wmma<<<B * 28 * 2, 32, 0, stream>>>(a0, c1_w, c1_b, a1);
    {
        int total = B * 6 * 14 * 14;
        k_avgpool2<<<(total + 255) / 256, 256, 0, stream>>>(a1, p1, total, 6, 14, 14);
    }
    k_conv3_wmma<<<B * 10, 32, 0, stream>>>(p1, c3_w, c3_b, a3);
    {
        int total = B * 16 * 5 * 5;
        k_avgpool2<<<(total + 255) / 256, 256, 0, stream>>>(a3, p3, total, 16, 5, 5);
    }
    k_c5_wmma<<<(B / 16) * 8, 32, 0, stream>>>(p3, c5_w, c5_b, a5);
    k_fc1_wmma<<<(B / 16) * 6, 32, 0, stream>>>(a5, l1_w, l1_b, a6);
    k_fc2_wmma<<<B / 16, 32, 0, stream>>>(a6, l2_w, l2_b, out);
}